// CFDAGCN_86122684219978
// MI455X (gfx1250) — compile-verified
//
#include <hip/hip_runtime.h>
#include <stdint.h>
#include <math.h>

typedef __attribute__((ext_vector_type(2))) float v2f;
typedef __attribute__((ext_vector_type(8))) float v8f;

#define SLD 1092  // row stride (floats) of ping-pong buffers: >= 1088 (=ceil64(1027)), even
#define BK  64    // K-chunk staged in LDS
#define SWS 68    // LDS row stride (floats): even (b64 align), 4n+k bank-spread

static __device__ __forceinline__ v8f wmma4(v2f a, v2f b, v8f c) {
  return __builtin_amdgcn_wmma_f32_16x16x4_f32(false, a, false, b, (short)0, c,
                                               false, false);
}

// ----------------------------- utility kernels ------------------------------

__global__ void k_zero_f32(float* __restrict__ p, long long n) {
  long long i = (long long)blockIdx.x * blockDim.x + threadIdx.x;
  if (i < n) p[i] = 0.0f;
}

__global__ void k_zero_i32(int* __restrict__ p, long long n) {
  long long i = (long long)blockIdx.x * blockDim.x + threadIdx.x;
  if (i < n) p[i] = 0;
}

// count in-degree at dst (edge indices are int64 in the reference)
__global__ void k_count_deg(const long long* __restrict__ dst, long long E,
                            int* __restrict__ deg) {
  long long e = (long long)blockIdx.x * blockDim.x + threadIdx.x;
  if (e < E) atomicAdd(&deg[(int)dst[e]], 1);
}

__global__ void k_dinv(const int* __restrict__ deg, float* __restrict__ dinv, int N) {
  int i = blockIdx.x * blockDim.x + threadIdx.x;
  if (i < N) dinv[i] = 1.0f / sqrtf((float)deg[i] + 1.0f);  // self-loop +1
}

// buf[n, 0:5] = x[n, 0:5]; buf[n,5] = sdf[n]
__global__ void k_build_in(const float* __restrict__ x, const float* __restrict__ sdf,
                           float* __restrict__ buf, int N) {
  int n = blockIdx.x * blockDim.x + threadIdx.x;
  if (n >= N) return;
  float* r = buf + (long long)n * SLD;
  const float* xr = x + (long long)n * 5;
  r[0] = xr[0]; r[1] = xr[1]; r[2] = xr[2]; r[3] = xr[3]; r[4] = xr[4];
  r[5] = sdf[n];
}

// zero buf[:, colOff : colOff+1024]
__global__ void k_zero_cols1024(float* __restrict__ buf, int colOff, long long total) {
  long long gid = (long long)blockIdx.x * blockDim.x + threadIdx.x;
  if (gid >= total) return;
  long long n = gid >> 10;
  int c = (int)(gid & 1023);
  buf[n * SLD + colOff + c] = 0.0f;
}

// ------------------------------- knn (k=3) ----------------------------------
__global__ __launch_bounds__(256) void k_knn(const float* __restrict__ x,
                                             const float* __restrict__ cpos,
                                             const float* __restrict__ cy,
                                             float* __restrict__ outbuf,
                                             int N, int NC) {
  __shared__ float sx[2048];
  __shared__ float sy[2048];
  for (int i = threadIdx.x; i < NC && i < 2048; i += 256) {
    sx[i] = cpos[2 * i + 0];
    sy[i] = cpos[2 * i + 1];
  }
  __syncthreads();
  int n = blockIdx.x * 256 + threadIdx.x;
  if (n >= N) return;
  float px = x[(long long)n * 5 + 0];
  float py = x[(long long)n * 5 + 1];
  float bd0 = 3.4e38f, bd1 = 3.4e38f, bd2 = 3.4e38f;
  int bi0 = 0, bi1 = 0, bi2 = 0;
  for (int j = 0; j < NC; ++j) {
    float dx = px - sx[j];
    float dy = py - sy[j];
    float d2 = dx * dx + dy * dy;
    if (d2 < bd0)      { bd2 = bd1; bi2 = bi1; bd1 = bd0; bi1 = bi0; bd0 = d2; bi0 = j; }
    else if (d2 < bd1) { bd2 = bd1; bi2 = bi1; bd1 = d2; bi1 = j; }
    else if (d2 < bd2) { bd2 = d2; bi2 = j; }
  }
  float w0 = 1.0f / fmaxf(bd0, 1e-16f);
  float w1 = 1.0f / fmaxf(bd1, 1e-16f);
  float w2 = 1.0f / fmaxf(bd2, 1e-16f);
  float wsum = w0 + w1 + w2;
  float* r = outbuf + (long long)n * SLD;
  #pragma unroll
  for (int c = 0; c < 3; ++c) {
    float num = cy[(long long)bi0 * 3 + c] * w0 +
                cy[(long long)bi1 * 3 + c] * w1 +
                cy[(long long)bi2 * 3 + c] * w2;
    r[c] = num / wsum;
  }
}

// ------------------------------ WMMA f32 GEMM -------------------------------
// C[M,Nout] = A[M,Kpad] @ W[K,Nout] via V_WMMA_F32_16X16X4_F32.
// Kpad is a multiple of 64 and A's columns [K, Kpad) are pre-zeroed, so the
// K loop is tail-free: every chunk runs the unmasked hot path
// (16 global b64 A-loads, 64 ds b64 B-loads, 64 WMMAs). The real K only masks
// the cold LDS fill of the W panel (zero-padded there, which also absorbs the
// Nout tail for the 3-wide final layer).
// Fragment layouts per ISA 7.12.2:
//   A 16x4: lanes 0-15: M=lane, vgprs K+0/K+1; lanes 16-31: M=lane-16, K+2/K+3.
//   B 4x16: lanes 0-15: N=lane, K+0/K+1;       lanes 16-31: N=lane-16, K+2/K+3.
//   C/D:    vgpr v -> M=v (lanes 0-15) / M=v+8 (lanes 16-31); N = lane & 15.
__global__ __launch_bounds__(256) void k_gemm_wmma(
    const float* __restrict__ A, int lda,
    const float* __restrict__ W, int ldw,
    float* __restrict__ C, int ldc,
    int M, int K, int Kpad, int Nout) {
  __shared__ float sW[64 * SWS];

  int tid = threadIdx.x;
  int wave = tid >> 5;
  int lane = tid & 31;
  int m0 = blockIdx.x * 128 + wave * 16;
  int n0 = blockIdx.y * 64;
  int lhalf = lane & 15;
  int khi = (lane >> 4) << 1;  // 0 or 2
  const float* Arow = A + (long long)(m0 + lhalf) * lda;

  // LDS-fill mapping: tid -> n = tid&63 (coalesced global read), k = 4p + tid>>6
  int fn = tid & 63;
  int fk0 = tid >> 6;
  int gn = n0 + fn;
  bool nok = (gn < Nout);
  float* sWr = sW + fn * SWS;

  v8f acc0 = {}, acc1 = {}, acc2 = {}, acc3 = {};

  const float* sB0 = sW + (lhalf + 0)  * SWS + khi;
  const float* sB1 = sW + (lhalf + 16) * SWS + khi;
  const float* sB2 = sW + (lhalf + 32) * SWS + khi;
  const float* sB3 = sW + (lhalf + 48) * SWS + khi;

  for (int k0 = 0; k0 < Kpad; k0 += BK) {
    __syncthreads();
    #pragma unroll
    for (int p = 0; p < 16; ++p) {
      int kk = p * 4 + fk0;
      float v = 0.0f;
      if (nok && (k0 + kk) < K) v = W[(long long)(k0 + kk) * ldw + gn];
      sWr[kk] = v;
    }
    __syncthreads();

    // prefetch next A chunk (speculative; lowers to global_prefetch_b8)
    __builtin_prefetch(Arow + k0 + BK, 0, 1);

    #pragma unroll
    for (int s = 0; s < 16; ++s) {
      v2f a  = *(const v2f*)(Arow + k0 + s * 4 + khi);
      v2f b0 = *(const v2f*)(sB0 + s * 4);
      v2f b1 = *(const v2f*)(sB1 + s * 4);
      v2f b2 = *(const v2f*)(sB2 + s * 4);
      v2f b3 = *(const v2f*)(sB3 + s * 4);
      acc0 = wmma4(a, b0, acc0);
      acc1 = wmma4(a, b1, acc1);
      acc2 = wmma4(a, b2, acc2);
      acc3 = wmma4(a, b3, acc3);
    }
  }

  int rbase = m0 + ((lane >> 4) ? 8 : 0);
  {
    int nn = n0 + 0 * 16 + lhalf;
    if (nn < Nout) {
      #pragma unroll
      for (int v = 0; v < 8; ++v) C[(long long)(rbase + v) * ldc + nn] = acc0[v];
    }
  }
  {
    int nn = n0 + 1 * 16 + lhalf;
    if (nn < Nout) {
      #pragma unroll
      for (int v = 0; v < 8; ++v) C[(long long)(rbase + v) * ldc + nn] = acc1[v];
    }
  }
  {
    int nn = n0 + 2 * 16 + lhalf;
    if (nn < Nout) {
      #pragma unroll
      for (int v = 0; v < 8; ++v) C[(long long)(rbase + v) * ldc + nn] = acc2[v];
    }
  }
  {
    int nn = n0 + 3 * 16 + lhalf;
    if (nn < Nout) {
      #pragma unroll
      for (int v = 0; v < 8; ++v) C[(long long)(rbase + v) * ldc + nn] = acc3[v];
    }
  }
}

// --------------------------- GCN edge aggregation ---------------------------
// F=512 specialization: one b128 H-load + 4 atomics per thread.
// OUT[dst, colOff+c..c+3] += H[src, c..c+3] * dinv[src]*dinv[dst]
__global__ void k_edge_scatter512(const float* __restrict__ H,
                                  const long long* __restrict__ src,
                                  const long long* __restrict__ dst,
                                  const float* __restrict__ dinv,
                                  float* __restrict__ OUT, int colOff,
                                  long long E) {
  long long gid = (long long)blockIdx.x * blockDim.x + threadIdx.x;
  long long total = E << 7;  // E * 128 four-float chunks
  if (gid >= total) return;
  long long e = gid >> 7;
  int c = (int)(gid & 127) << 2;
  int s = (int)src[e];
  int d = (int)dst[e];
  float nrm = dinv[s] * dinv[d];
  const float4 h = *(const float4*)(H + ((long long)s << 9) + c);
  float* o = OUT + (long long)d * SLD + colOff + c;
  atomicAdd(o + 0, h.x * nrm);
  atomicAdd(o + 1, h.y * nrm);
  atomicAdd(o + 2, h.z * nrm);
  atomicAdd(o + 3, h.w * nrm);
}

// generic (used for the 3-wide final layer)
__global__ void k_edge_scatter_gen(const float* __restrict__ H, int ldh, int F,
                                   const long long* __restrict__ src,
                                   const long long* __restrict__ dst,
                                   const float* __restrict__ dinv,
                                   float* __restrict__ OUT, int ldo, int colOff,
                                   long long E) {
  long long gid = (long long)blockIdx.x * blockDim.x + threadIdx.x;
  long long total = E * (long long)F;
  if (gid >= total) return;
  long long e = gid / F;
  int c = (int)(gid - e * (long long)F);
  int s = (int)src[e];
  int d = (int)dst[e];
  float nrm = dinv[s] * dinv[d];
  float val = H[(long long)s * ldh + c] * nrm;
  atomicAdd(&OUT[(long long)d * ldo + colOff + c], val);
}

// OUT += self-loop (H * dinv^2) + bias, optional relu  (F = 512)
__global__ void k_finish512(float* __restrict__ OUT, int colOff,
                            const float* __restrict__ H,
                            const float* __restrict__ dinv,
                            const float* __restrict__ bias, int relu,
                            long long total) {
  long long gid = (long long)blockIdx.x * blockDim.x + threadIdx.x;
  if (gid >= total) return;
  long long n = gid >> 9;
  int c = (int)(gid & 511);
  float di = dinv[n];
  long long o = n * SLD + colOff + c;
  float v = OUT[o] + H[(n << 9) + c] * di * di + bias[c];
  if (relu) v = fmaxf(v, 0.0f);
  OUT[o] = v;
}

// generic (3-wide final layer; writes d_out)
__global__ void k_finish_gen(float* __restrict__ OUT, int ldo, int colOff,
                             const float* __restrict__ H, int ldh, int F,
                             const float* __restrict__ dinv,
                             const float* __restrict__ bias, int relu, int N) {
  long long gid = (long long)blockIdx.x * blockDim.x + threadIdx.x;
  long long total = (long long)N * F;
  if (gid >= total) return;
  long long n = gid / F;
  int c = (int)(gid - n * (long long)F);
  float di = dinv[n];
  long long o = n * ldo + colOff + c;
  float v = OUT[o] + H[n * (long long)ldh + c] * di * di + bias[c];
  if (relu) v = fmaxf(v, 0.0f);
  OUT[o] = v;
}

// --------------------------------- driver -----------------------------------

extern "C" void kernel_launch(void* const* d_in, const int* in_sizes, int n_in,
                              void* d_out, int out_size, void* d_ws, size_t ws_size,
                              hipStream_t stream) {
  const float* x      = (const float*)d_in[0];
  const float* sdf    = (const float*)d_in[1];
  const float* cpos   = (const float*)d_in[2];
  const float* cy     = (const float*)d_in[3];
  const float* W_pre0 = (const float*)d_in[4];
  const float* b_pre0 = (const float*)d_in[5];
  const float* W_pre1 = (const float*)d_in[6];
  const float* b_pre1 = (const float*)d_in[7];
  const float* W_pre2 = (const float*)d_in[8];
  const float* b_pre2 = (const float*)d_in[9];
  const float* W_end0 = (const float*)d_in[10];
  const float* b_end0 = (const float*)d_in[11];
  const float* W_end1 = (const float*)d_in[12];
  const float* b_end1 = (const float*)d_in[13];
  const float* W_end2 = (const float*)d_in[14];
  const float* b_end2 = (const float*)d_in[15];
  const long long* e1 = (const long long*)d_in[16];
  const long long* e2 = (const long long*)d_in[17];

  const int N  = in_sizes[1];           // 16384
  const int NC = in_sizes[2] / 2;       // 2048
  const long long E1 = in_sizes[16] / 2;
  const long long E2 = in_sizes[17] / 2;
  const long long* s1 = e1;
  const long long* d1 = e1 + E1;
  const long long* s2 = e2;
  const long long* d2 = e2 + E2;
  const int HID = 512;

  float* bufA  = (float*)d_ws;
  float* bufB  = bufA + (size_t)N * SLD;
  float* Hbuf  = bufB + (size_t)N * SLD;
  float* dinv1 = Hbuf + (size_t)N * HID;
  float* dinv2 = dinv1 + N;
  int*   deg   = (int*)(dinv2 + N);
  (void)ws_size; (void)n_in; (void)out_size;

  float* out = (float*)d_out;

  auto cdiv = [](long long a, long long b) { return (unsigned)((a + b - 1) / b); };

  // full zero of ping-pong buffers: establishes the zero K-padding columns
  k_zero_f32<<<cdiv((long long)N * SLD, 256), 256, 0, stream>>>(
      bufA, (long long)N * SLD);
  k_zero_f32<<<cdiv((long long)N * SLD, 256), 256, 0, stream>>>(
      bufB, (long long)N * SLD);

  // symmetric-normalization coefficients for both adjacencies
  k_zero_i32<<<cdiv(N, 256), 256, 0, stream>>>(deg, N);
  k_count_deg<<<cdiv(E1, 256), 256, 0, stream>>>(d1, E1, deg);
  k_dinv<<<cdiv(N, 256), 256, 0, stream>>>(deg, dinv1, N);
  k_zero_i32<<<cdiv(N, 256), 256, 0, stream>>>(deg, N);
  k_count_deg<<<cdiv(E2, 256), 256, 0, stream>>>(d2, E2, deg);
  k_dinv<<<cdiv(N, 256), 256, 0, stream>>>(deg, dinv2, N);

  // input features: bufA[:,0:6] = [x, sdf]  (cols 6..63 stay zero)
  k_build_in<<<cdiv(N, 256), 256, 0, stream>>>(x, sdf, bufA, N);

  // One shared-weight GCN over both adjacencies, halves at colOff / colOff+512.
  auto dual = [&](const float* IN, int Kdim, int Kpad, const float* Wt,
                  const float* bt, float* OUT, int colOff, int relu) {
    dim3 gg((unsigned)(N / 128), (unsigned)((HID + 63) / 64));
    k_gemm_wmma<<<gg, 256, 0, stream>>>(IN, SLD, Wt, HID, Hbuf, HID,
                                        N, Kdim, Kpad, HID);
    k_zero_cols1024<<<cdiv((long long)N * 1024, 256), 256, 0, stream>>>(
        OUT, colOff, (long long)N * 1024);
    k_edge_scatter512<<<cdiv(E1 << 7, 256), 256, 0, stream>>>(
        Hbuf, s1, d1, dinv1, OUT, colOff, E1);
    k_edge_scatter512<<<cdiv(E2 << 7, 256), 256, 0, stream>>>(
        Hbuf, s2, d2, dinv2, OUT, colOff + HID, E2);
    k_finish512<<<cdiv((long long)N * HID, 256), 256, 0, stream>>>(
        OUT, colOff, Hbuf, dinv1, bt, relu, (long long)N * HID);
    k_finish512<<<cdiv((long long)N * HID, 256), 256, 0, stream>>>(
        OUT, colOff + HID, Hbuf, dinv2, bt, relu, (long long)N * HID);
  };

  // pre0: [x|sdf] K=6 (pad 64) -> bufB[:,0:1024]
  dual(bufA, 6, 64, W_pre0, b_pre0, bufB, 0, 1);
  // pre1: bufB K=1024 -> bufA[:,0:1024]
  dual(bufB, 1024, 1024, W_pre1, b_pre1, bufA, 0, 1);
  // knn interpolate -> bufB[:,0:3] (pre0's output in bufB is dead after pre1)
  k_knn<<<cdiv(N, 256), 256, 0, stream>>>(x, cpos, cy, bufB, N, NC);
  // pre2: bufA K=1024 -> bufB[:,3:1027]  (cols 0..2 hold fine_y; 1027.. stay 0)
  dual(bufA, 1024, 1024, W_pre2, b_pre2, bufB, 3, 1);
  // end0: bufB[:,0:1027] K=1027 (pad 1088; cols 1027..1087 are zero) -> bufA
  dual(bufB, 1027, 1088, W_end0, b_end0, bufA, 0, 1);
  // end1: bufA K=1024 -> bufB[:,0:1024]
  dual(bufA, 1024, 1024, W_end1, b_end1, bufB, 0, 1);

  // end2: single adjacency A1, out_dim=3, no relu, into d_out
  dim3 g2((unsigned)(N / 128), 1);
  k_gemm_wmma<<<g2, 256, 0, stream>>>(bufB, SLD, W_end2, 3, Hbuf, HID,
                                      N, 1024, 1024, 3);
  k_zero_f32<<<cdiv((long long)N * 3, 256), 256, 0, stream>>>(out, (long long)N * 3);
  k_edge_scatter_gen<<<cdiv(E1 * 3, 256), 256, 0, stream>>>(
      Hbuf, HID, 3, s1, d1, dinv1, out, 3, 0, E1);
  k_finish_gen<<<cdiv((long long)N * 3, 256), 256, 0, stream>>>(
      out, 3, 0, Hbuf, HID, 3, dinv1, b_end2, 0, N);
}